// NeuralODE_46935402611059
// MI455X (gfx1250) — compile-verified
//
#include <hip/hip_runtime.h>
#include <math.h>

#define DD 32
#define HH 128
#define BATCH 32
#define NSTEPS 48
#define T1STRIDE 132   // even (8B-aligned b64 loads), mod 64 == 4 -> conflict-free banks

typedef __attribute__((ext_vector_type(2))) float v2f;
typedef __attribute__((ext_vector_type(8))) float v8f;

// Dormand-Prince coefficients
__constant__ float dA[6][6] = {
  {0.2f, 0.f, 0.f, 0.f, 0.f, 0.f},
  {3.f/40.f, 9.f/40.f, 0.f, 0.f, 0.f, 0.f},
  {44.f/45.f, -56.f/15.f, 32.f/9.f, 0.f, 0.f, 0.f},
  {19372.f/6561.f, -25360.f/2187.f, 64448.f/6561.f, -212.f/729.f, 0.f, 0.f},
  {9017.f/3168.f, -355.f/33.f, 46732.f/5247.f, 49.f/176.f, -5103.f/18656.f, 0.f},
  {35.f/384.f, 0.f, 500.f/1113.f, 125.f/192.f, -2187.f/6784.f, 11.f/84.f}};
__constant__ float dC[7]  = {0.f, 0.2f, 0.3f, 0.8f, 8.f/9.f, 1.f, 1.f};
__constant__ float dB5[7] = {35.f/384.f, 0.f, 500.f/1113.f, 125.f/192.f,
                             -2187.f/6784.f, 11.f/84.f, 0.f};
__constant__ float dE[7]  = {35.f/384.f - 5179.f/57600.f, 0.f,
                             500.f/1113.f - 7571.f/16695.f,
                             125.f/192.f - 393.f/640.f,
                             -2187.f/6784.f + 92097.f/339200.f,
                             11.f/84.f - 187.f/2100.f, -1.f/40.f};

__global__ __launch_bounds__(256)
void node_cnf_kernel(const float* __restrict__ x,
                     const float* __restrict__ W1, const float* __restrict__ b1,
                     const float* __restrict__ W2, const float* __restrict__ b2,
                     const float* __restrict__ W3, const float* __restrict__ b3,
                     float* __restrict__ out)
{
  __shared__ float sW1[HH * (DD + 1)];        // 128 x 33 row-major
  __shared__ float sW3[DD * HH];              // 32 x 128 row-major
  __shared__ float sT1T[(DD + 1) * T1STRIDE]; // transposed [T1 | h1]: row c holds K=0..127
  __shared__ float sb1[HH], sb2[HH], sb3[DD];
  __shared__ float ss1[HH], sh2pre[HH], sh2[HH], ss2[HH];
  __shared__ float sz[DD + 2], sztmp[DD + 2], sz5[DD + 2];
  __shared__ float sks[7][DD + 2];
  __shared__ float sred[256];                 // dxdt partials [seg][row]
  __shared__ float swred[8];                  // per-wave scalar partials
  __shared__ float sctl[4];                   // 0:t  1:dt  2:done
  __shared__ float sbr[2];                    // 0:doUpdateZ

  const int tid  = (int)threadIdx.x;
  const int lane = tid & 31;
  const int wv   = tid >> 5;                  // wave 0..7 -> M-tile of U2
  const int b    = (int)blockIdx.x;
  const int bcol = lane & 15;                 // N within tile / A row select
  const int koff = (lane >> 4) * 2;           // K sub-offset per ISA A/B layout
  const int rbase = 16 * wv + 8 * (lane >> 4);
  const int kT   = tid & 127;                 // T1T build: K row handled by this thread
  const int cbase = (tid >> 7) * 16;          // T1T build: first of 16 consecutive columns

  // ---- stage weights into LDS (W2 stays in registers only) ----
  for (int i = tid; i < HH * (DD + 1); i += 256) sW1[i] = W1[i];
  for (int i = tid; i < DD * HH;       i += 256) sW3[i] = W3[i];
  if (tid < HH) { sb1[tid] = b1[tid]; sb2[tid] = b2[tid]; }
  if (tid < DD) sb3[tid] = b3[tid];

  // ---- preload W2 A-matrix fragments (rows 16*wv..16*wv+15) into VGPRs ----
  const int arow = 16 * wv + bcol;
  v2f afrag[32];
#pragma unroll
  for (int kk = 0; kk < 32; ++kk) {
    const float* p = W2 + arow * HH + 4 * kk + koff;
    afrag[kk].x = p[0];
    afrag[kk].y = p[1];
  }

  // ---- init ODE state ----
  if (tid < DD) sz[tid] = x[b * DD + tid];
  if (tid == 0) { sz[DD] = 0.f; sctl[0] = 0.f; sctl[1] = 1e-4f; sctl[2] = 0.f; }
  __syncthreads();

  // ================= augmented RHS:  kout = f(t, zz) =================
  auto f_eval = [&](float t, const float* zz, float* kout) {
    const float t_eff = 1.0f - t;
    // layer 1 forward: h1 = tanh(W1 @ [x, t_eff] + b1); h1 goes straight to row 32
    if (tid < HH) {
      float acc = sb1[tid];
      const float* wr = &sW1[tid * (DD + 1)];
#pragma unroll
      for (int i = 0; i < DD; ++i) acc += wr[i] * zz[i];
      acc += wr[DD] * t_eff;
      float h = tanhf(acc);
      sT1T[DD * T1STRIDE + tid] = h;
      ss1[tid] = 1.f - h * h;
    }
    __syncthreads();

    // build transposed tangent matrix: sT1T[c][k] = s1[k]*W1[k][c]
    // fully unrolled, static offsets: thread = (k, 16-column slab)
    {
      const float s1v = ss1[kT];
      const float* wsrc = &sW1[kT * (DD + 1) + cbase];  // 16 consecutive floats
      float* wdst = &sT1T[cbase * T1STRIDE + kT];       // stride T1STRIDE per column
#pragma unroll
      for (int it = 0; it < 16; ++it)
        wdst[it * T1STRIDE] = s1v * wsrc[it];
    }
    __syncthreads();

    // U2 = W2 @ T1 via fp32 WMMA (2 tile chains); W2@h1 matvec rides the same
    // loop on register-resident afrag with one v_pk_fma_f32 per chunk.
    v8f c0 = {}; v8f c1 = {};
    v2f accv = {0.f, 0.f};
#pragma unroll
    for (int kk = 0; kk < 32; ++kk) {
      const int k0 = 4 * kk + koff;
      const v2f bf0 = *(const v2f*)&sT1T[bcol * T1STRIDE + k0];
      const v2f bf1 = *(const v2f*)&sT1T[(16 + bcol) * T1STRIDE + k0];
      const v2f bh  = *(const v2f*)&sT1T[DD * T1STRIDE + k0];
      c0 = __builtin_amdgcn_wmma_f32_16x16x4_f32(false, afrag[kk], false, bf0,
                                                 (short)0, c0, false, false);
      c1 = __builtin_amdgcn_wmma_f32_16x16x4_f32(false, afrag[kk], false, bf1,
                                                 (short)0, c1, false, false);
      accv += afrag[kk] * bh;
    }
    // finish layer-2 forward matvec: combine K-halves (lane l with lane l+16)
    float a2 = accv.x + accv.y;
    a2 += __shfl_xor(a2, 16, 32);
    if (lane < 16) sh2pre[16 * wv + bcol] = a2;
    __syncthreads();
    if (tid < HH) {
      float h = tanhf(sh2pre[tid] + sb2[tid]);
      sh2[tid] = h; ss2[tid] = 1.f - h * h;
    }
    __syncthreads();

    // trace(J) = sum_{i,k} W3[i,k] * s2[k] * U2[k,i]  (straight from accumulators)
    float part = 0.f;
#pragma unroll
    for (int r = 0; r < 8; ++r) {
      const int krow = rbase + r;
      const float sk = ss2[krow];
      part += sW3[bcol * HH + krow]        * sk * c0[r];
      part += sW3[(16 + bcol) * HH + krow] * sk * c1[r];
    }
#pragma unroll
    for (int off = 16; off > 0; off >>= 1) part += __shfl_xor(part, off, 32);
    if (lane == 0) swred[wv] = part;

    // dxdt = W3 @ h2 + b3, split 8 ways: thread = (seg, row)
    {
      const int row = tid & 31, seg = tid >> 5;
      float acc = 0.f;
      const float* wr = &sW3[row * HH + seg * 16];
      const float* hh = &sh2[seg * 16];
#pragma unroll
      for (int i = 0; i < 16; ++i) acc += wr[i] * hh[i];
      sred[seg * 32 + row] = acc;
    }
    __syncthreads();
    if (tid < DD) {
      float acc = sb3[tid];
#pragma unroll
      for (int s = 0; s < 8; ++s) acc += sred[s * 32 + tid];
      kout[tid] = -acc;
    }
    if (tid == 0) {
      float tr = 0.f;
#pragma unroll
      for (int w = 0; w < 8; ++w) tr += swred[w];
      kout[DD] = -tr;              // dlogp/dt = -trace
    }
    __syncthreads();
  };

  // ================= adaptive RK45 driver (block-uniform) =================
#pragma unroll 1
  for (int step = 0; step < NSTEPS; ++step) {
    const float t  = sctl[0];
    const float dt = sctl[1];
    const bool done = (sctl[2] != 0.f);
    const float dt_c = fminf(dt, 1.0f - t);

    f_eval(t, sz, sks[0]);
#pragma unroll 1
    for (int i = 0; i < 6; ++i) {
      if (tid <= DD) {
        float acc = 0.f;
        for (int m = 0; m <= i; ++m) acc += dA[i][m] * sks[m][tid];
        sztmp[tid] = sz[tid] + dt_c * acc;
      }
      __syncthreads();
      f_eval(t + dC[i + 1] * dt_c, sztmp, sks[i + 1]);
    }

    // embedded error estimate and step-size controller
    float term = 0.f;
    if (tid <= DD) {
      float a5 = 0.f, ae = 0.f;
      for (int m = 0; m < 7; ++m) {
        a5 += dB5[m] * sks[m][tid];
        ae += dE[m]  * sks[m][tid];
      }
      const float z5  = sz[tid] + dt_c * a5;
      const float err = dt_c * ae;
      sz5[tid] = z5;
      const float scale = 1e-5f + 1e-3f * fmaxf(fabsf(sz[tid]), fabsf(z5));
      const float q = err / scale;
      term = q * q;
    }
#pragma unroll
    for (int off = 16; off > 0; off >>= 1) term += __shfl_xor(term, off, 32);
    if (lane == 0) swred[wv] = term;
    __syncthreads();
    if (tid == 0) {
      float ssum = 0.f;
#pragma unroll
      for (int w = 0; w < 8; ++w) ssum += swred[w];
      const float ratio = sqrtf(ssum / (float)(DD + 1));
      const bool accept = (ratio <= 1.0f);
      float factor = 0.9f * powf(fmaxf(ratio, 1e-10f), -0.2f);
      factor = fminf(fmaxf(factor, 0.2f), 10.f);
      const float t_n  = accept ? (t + dt_c) : t;
      const float dt_n = dt_c * factor;
      if (!done) { sctl[0] = t_n; sctl[1] = dt_n; }
      if (t_n >= 1.0f - 1e-10f) sctl[2] = 1.f;
      sbr[0] = (accept && !done) ? 1.f : 0.f;
    }
    __syncthreads();
    if (sbr[0] != 0.f && tid <= DD) sz[tid] = sz5[tid];
    __syncthreads();
  }

  if (tid < DD) out[b * DD + tid] = sz[tid];
  if (tid == 0) out[BATCH * DD + b] = sz[DD];
}

extern "C" void kernel_launch(void* const* d_in, const int* in_sizes, int n_in,
                              void* d_out, int out_size, void* d_ws, size_t ws_size,
                              hipStream_t stream) {
  (void)in_sizes; (void)n_in; (void)out_size; (void)d_ws; (void)ws_size;
  const float* x  = (const float*)d_in[0];
  const float* W1 = (const float*)d_in[1];
  const float* b1 = (const float*)d_in[2];
  const float* W2 = (const float*)d_in[3];
  const float* b2 = (const float*)d_in[4];
  const float* W3 = (const float*)d_in[5];
  const float* b3 = (const float*)d_in[6];
  float* out = (float*)d_out;
  node_cnf_kernel<<<dim3(BATCH), dim3(256), 0, stream>>>(x, W1, b1, W2, b2, W3, b3, out);
}